// BayesLinearEMP_67808943669256
// MI455X (gfx1250) — compile-verified
//
#include <hip/hip_runtime.h>

typedef __attribute__((ext_vector_type(2))) float v2f;
typedef __attribute__((ext_vector_type(4))) float v4f;
typedef __attribute__((ext_vector_type(8))) float v8f;
typedef __attribute__((ext_vector_type(4))) int   v4i;

constexpr int Bsz = 128;   // batch
constexpr int Ich = 2048;  // input features (K)
constexpr int Och = 2048;  // output features (N-global)
constexpr int Md  = 20;    // modes
constexpr int KC  = 128;   // K chunk staged through LDS
constexpr int NW  = 8;     // waves per block (each owns 16 output cols)
constexpr int LSTR = KC + 4;      // LDS row stride: +4 floats -> conflict-free b64 frag reads
constexpr int CHUNKS = Bsz / 16;  // max 16-sample chunks per mode

#if defined(__HIP_DEVICE_COMPILE__) && __has_builtin(__builtin_amdgcn_global_load_async_to_lds_b128)
#define HAS_ASYNC_LDS 1
#else
#define HAS_ASYNC_LDS 0
#endif

// builtin signature (from hipcc diagnostic): (AS1 v4i*, AS3 v4i*, imm int, imm int)
typedef __attribute__((address_space(1))) v4i gv4i;
typedef __attribute__((address_space(3))) v4i lv4i;

__device__ __forceinline__ gv4i* gcast(const void* p) {
  return (gv4i*)(uintptr_t)p;
}
__device__ __forceinline__ lv4i* lcast(void* p) {
  // generic pointer to a __shared__ object: low 32 bits are the LDS offset
  return (lv4i*)(unsigned)(uintptr_t)p;
}
__device__ __forceinline__ void wait_async0() {
#if defined(__HIP_DEVICE_COMPILE__)
#if __has_builtin(__builtin_amdgcn_s_wait_asynccnt)
  __builtin_amdgcn_s_wait_asynccnt(0);
#else
  asm volatile("s_wait_asynccnt 0x0" ::: "memory");
#endif
#endif
}

// ---------------------------------------------------------------------------
// Kernel 1: deterministic grouping of samples by mode (tiny; 1 wave).
// lists[m*128 + j] = j-th sample (ascending) with mode m; counts[m] = group size
// ---------------------------------------------------------------------------
__global__ void bayes_group(const int* __restrict__ mode_idx,
                            int* __restrict__ counts,
                            int* __restrict__ lists) {
  int m = threadIdx.x;
  if (m >= Md) return;
  int c = 0;
  for (int b = 0; b < Bsz; ++b) {
    if (mode_idx[b] == m) lists[m * Bsz + c++] = b;
  }
  counts[m] = c;
}

// ---------------------------------------------------------------------------
// Kernel 2: per-mode grouped GEMM via V_WMMA_F32_16X16X4_F32, with
// GLOBAL_LOAD_ASYNC_TO_LDS_B128 staging of both x and W tiles.
// block = 256 threads (8 waves). blockIdx.x -> 128-wide output-column block,
// blockIdx.y -> (mode, 16-sample chunk). Each wave computes a 16x16 tile
// (16 samples x 16 output cols) over the full K=2048 reduction.
// ---------------------------------------------------------------------------
__launch_bounds__(256, 1)
__global__ void bayes_wmma(const float* __restrict__ x,
                           const float* __restrict__ w,
                           const float* __restrict__ bias,
                           const int*  __restrict__ counts,
                           const int*  __restrict__ lists,
                           float* __restrict__ out) {
  const int oblk  = blockIdx.x;             // 0..15 : block of 128 output cols
  const int m     = blockIdx.y / CHUNKS;    // mode
  const int chunk = blockIdx.y % CHUNKS;    // 16-sample chunk within mode group
  const int cnt   = counts[m];
  if (chunk * 16 >= cnt) return;            // uniform early exit (empty chunk)
  const int valid = min(16, cnt - chunk * 16);

  __shared__ float xs[16][LSTR];            // x tile: 16 samples x KC
  __shared__ float wt[NW][16][LSTR];        // per-wave W tile: 16 out-rows x KC
  __shared__ int   sl[16];                  // sample list for this chunk

  const int t    = threadIdx.x;
  const int wave = t >> 5;
  const int lane = t & 31;
  const int rr   = lane & 15;               // A: M row / B: N col / D: N col
  const int hh   = lane >> 4;               // K-half selector

  if (t < 16) sl[t] = (t < valid) ? lists[m * Bsz + chunk * 16 + t] : -1;
  __syncthreads();

  const int  xr = t >> 4;                   // this thread's x-staging row
  const int  xc = (t & 15) * 4;             // 2 float4 per thread covers KC=128
  const int  xsmp = sl[xr];                 // sample id (or -1 for padding)

#if HAS_ASYNC_LDS
  // Padded rows are never async-overwritten: zero them once up front.
  if (xsmp < 0) {
    *(v4f*)&xs[xr][xc]      = (v4f){};
    *(v4f*)&xs[xr][xc + 64] = (v4f){};
  }
#endif

  const int  ocol0 = oblk * 128 + wave * 16;          // wave's 16 output cols
  const long wbase = (long)m * Och * Ich;

  v8f acc = {};

  for (int ic = 0; ic < Ich; ic += KC) {
#if HAS_ASYNC_LDS
    // ---- async-stage x tile [16 x KC] (ASYNCcnt; no VGPR round-trip) ----
    if (xsmp >= 0) {
      const float* p = x + (long)xsmp * Ich + ic;
      __builtin_amdgcn_global_load_async_to_lds_b128(
          gcast(p + xc),      lcast(&xs[xr][xc]),      0, 0);
      __builtin_amdgcn_global_load_async_to_lds_b128(
          gcast(p + xc + 64), lcast(&xs[xr][xc + 64]), 0, 0);
    }
    // ---- async-stage this wave's W tile [16 x KC]: full-wave 512B rows ----
    {
      const float* wp = w + wbase + (long)ocol0 * Ich + ic;
      #pragma unroll
      for (int r = 0; r < 16; ++r) {
        __builtin_amdgcn_global_load_async_to_lds_b128(
            gcast(wp + (long)r * Ich + lane * 4),
            lcast(&wt[wave][r][lane * 4]), 0, 0);
      }
    }
    wait_async0();
    __syncthreads();
#else
    // ---- synchronous fallback staging through VGPRs ----
    {
      v4f v0 = {}, v1 = {};
      if (xsmp >= 0) {
        const float* p = x + (long)xsmp * Ich + ic;
        v0 = *(const v4f*)(p + xc);
        v1 = *(const v4f*)(p + xc + 64);
      }
      *(v4f*)&xs[xr][xc]      = v0;
      *(v4f*)&xs[xr][xc + 64] = v1;
    }
    {
      const float* wp = w + wbase + (long)ocol0 * Ich + ic;
      #pragma unroll
      for (int r = 0; r < 16; ++r) {
        // weights are streamed once per mode-chunk: non-temporal keeps L2 for x
        v4f v = __builtin_nontemporal_load((const v4f*)(wp + (long)r * Ich + lane * 4));
        *(v4f*)&wt[wave][r][lane * 4] = v;
      }
    }
    __syncthreads();
#endif

    // ---- 32 WMMA steps of K=4 each ----
    // A lane layout: M = lane&15, K = vgpr + 2*(lane>=16)  (ISA 16x4 f32 table)
    // B mirrored   : N = lane&15, K = vgpr + 2*(lane>=16); B[k][n] = W[ocol0+n][ic+k]
    #pragma unroll
    for (int kk = 0; kk < KC; kk += 4) {
      v2f a = *(const v2f*)&xs[rr][kk + 2 * hh];
      v2f b = *(const v2f*)&wt[wave][rr][kk + 2 * hh];
      acc = __builtin_amdgcn_wmma_f32_16x16x4_f32(
          /*neg_a=*/false, a, /*neg_b=*/false, b,
          /*c_mod=*/(short)0, acc, /*reuse_a=*/false, /*reuse_b=*/false);
    }
    __syncthreads();
  }

  // ---- epilogue: D layout lane -> N = lane&15, M = 8*(lane>=16)+v ----
  const int   ocol = ocol0 + rr;
  const float bv   = bias[(long)m * Och + ocol];
  #pragma unroll
  for (int v = 0; v < 8; ++v) {
    const int ml = hh * 8 + v;              // local sample row of acc[v]
    if (ml < valid) {
      const int b = sl[ml];
      out[(long)b * Och + ocol] = acc[v] + bv;
    }
  }
}

// ---------------------------------------------------------------------------
extern "C" void kernel_launch(void* const* d_in, const int* in_sizes, int n_in,
                              void* d_out, int out_size, void* d_ws, size_t ws_size,
                              hipStream_t stream) {
  const float* x    = (const float*)d_in[0];   // [128, 2048] f32
  const float* w    = (const float*)d_in[1];   // [20, 2048, 2048] f32
  const float* bias = (const float*)d_in[2];   // [20, 2048] f32
  const int*   midx = (const int*)d_in[3];     // [128] i32
  float*       out  = (float*)d_out;           // [128, 2048] f32

  int* counts = (int*)d_ws;        // 20 ints (padded to 32)
  int* lists  = counts + 32;       // 20 * 128 ints

  bayes_group<<<1, 32, 0, stream>>>(midx, counts, lists);

  dim3 grid(Och / (NW * 16), Md * CHUNKS);   // (16, 160)
  bayes_wmma<<<grid, 256, 0, stream>>>(x, w, bias, counts, lists, out);
}